// ConcatReadModel_34583076667914
// MI455X (gfx1250) — compile-verified
//
#include <hip/hip_runtime.h>

// ---------------------------------------------------------------------------
// MI455X (gfx1250) implementation.
//
// Phase 1 (ff_tokens_kernel): token-parallel embed->FF->LN->key projections
//   using native fp32 WMMA (v_wmma_f32_16x16x4_f32), 320 WMMA per 16-token
//   tile, weights pre-packed so every B-fragment is one 8B load.
// Phase 2 (scan_kernel): per-batch sequential delta-rule scan; one wave per
//   (batch, memory), lane = row of the 32x32 fast-weight matrix held in VGPRs;
//   k_t is wave-uniform -> scalar-load friendly. Fused final readout GEMV.
// ---------------------------------------------------------------------------

typedef __attribute__((ext_vector_type(2))) float v2f;
typedef __attribute__((ext_vector_type(8))) float v8f;

#define HID   64
#define FFD   128
#define HALFD 32
#define LSEQ  2048
#define BSZ   256

static __device__ inline v8f wmma_f32(v2f a, v2f b, v8f c) {
  // (neg_a, A, neg_b, B, c_mod, C, reuse_a, reuse_b)
  return __builtin_amdgcn_wmma_f32_16x16x4_f32(false, a, false, b, (short)0, c,
                                               false, false);
}

// Pack row-major W[K][N] into Wp[(K/2)][N][2] so that a B-fragment
// {W[K0][n], W[K0+1][n]} is a contiguous float2.
__global__ void pack_pairs_kernel(const float* __restrict__ src,
                                  float* __restrict__ dst, int K2, int N) {
  int idx = blockIdx.x * blockDim.x + threadIdx.x;
  if (idx >= K2 * N) return;
  int k2 = idx / N, n = idx % N;
  dst[2 * idx + 0] = src[(2 * k2 + 0) * N + n];
  dst[2 * idx + 1] = src[(2 * k2 + 1) * N + n];
}

// Phase 1: 4 waves per block, one 16-token tile per wave.
// Dynamic LDS: per wave 3072 floats (aBuf 16x128, eBuf 16x64; hBuf aliases aBuf).
__global__ void __launch_bounds__(128)
ff_tokens_kernel(const int* __restrict__ seq, const float* __restrict__ embed,
                 const float* __restrict__ W1p, const float* __restrict__ b1,
                 const float* __restrict__ W2p, const float* __restrict__ b2,
                 const float* __restrict__ gamma, const float* __restrict__ beta,
                 const float* __restrict__ Wsp, const float* __restrict__ bs,
                 const float* __restrict__ Wep, const float* __restrict__ be,
                 float* __restrict__ ksb, float* __restrict__ keb) {
  extern __shared__ float smem[];
  const int tid  = threadIdx.x;
  const int wave = tid >> 5;
  const int lane = tid & 31;
  const int lr   = lane & 15;  // A-frag row / C-frag column-within-tile
  const int hi   = lane >> 4;  // half selector

  float* aBuf = smem + wave * 3072;  // 16 x 128 intermediate
  float* eBuf = aBuf + 2048;         // 16 x 64 embeddings
  float* hBuf = aBuf;                // 16 x 64 hidden (aliases aBuf after FF2)

  const long gt = (long)blockIdx.x * 4 + wave;  // tile id, L/16 = 128 per batch
  const int b  = (int)(gt >> 7);
  const int t0 = ((int)gt & 127) << 4;

  // --- load e as A-fragments, stash rows into eBuf ---
  const int tok = seq[(size_t)b * LSEQ + t0 + lr];
  const float* erow = embed + (size_t)tok * HID;
  v2f ea[16];
#pragma unroll
  for (int kc = 0; kc < 16; ++kc) {
    const int k0 = 4 * kc + 2 * hi;
    v2f v;
    v.x = erow[k0];
    v.y = erow[k0 + 1];
    ea[kc] = v;
    *(v2f*)&eBuf[lr * HID + k0] = v;
  }

  // --- FF1: (16x64) @ (64x128) + b1, via 16 K-chunks x 8 N-tiles WMMA ---
  v8f acc1[8];
#pragma unroll
  for (int nt = 0; nt < 8; ++nt) {
    float bv = b1[nt * 16 + lr];
#pragma unroll
    for (int r = 0; r < 8; ++r) acc1[nt][r] = bv;
  }
#pragma unroll
  for (int kc = 0; kc < 16; ++kc) {
    const int k2 = 2 * kc + hi;
#pragma unroll
    for (int nt = 0; nt < 8; ++nt) {
      v2f bf = *(const v2f*)&W1p[((size_t)k2 * FFD + nt * 16 + lr) * 2];
      acc1[nt] = wmma_f32(ea[kc], bf, acc1[nt]);
    }
  }

  // relu, spill C-layout -> row-major LDS for transpose to A-layout
#pragma unroll
  for (int nt = 0; nt < 8; ++nt)
#pragma unroll
    for (int r = 0; r < 8; ++r) {
      float v = acc1[nt][r];
      aBuf[(r + 8 * hi) * FFD + nt * 16 + lr] = v > 0.f ? v : 0.f;
    }
  __syncthreads();

  // --- FF2: (16x128) @ (128x64) + b2 ---
  v8f acc2[4];
#pragma unroll
  for (int nt = 0; nt < 4; ++nt) {
    float bv = b2[nt * 16 + lr];
#pragma unroll
    for (int r = 0; r < 8; ++r) acc2[nt][r] = bv;
  }
#pragma unroll
  for (int kc = 0; kc < 32; ++kc) {
    v2f af = *(const v2f*)&aBuf[lr * FFD + 4 * kc + 2 * hi];
    const int k2 = 2 * kc + hi;
#pragma unroll
    for (int nt = 0; nt < 4; ++nt) {
      v2f bf = *(const v2f*)&W2p[((size_t)k2 * HID + nt * 16 + lr) * 2];
      acc2[nt] = wmma_f32(af, bf, acc2[nt]);
    }
  }
  __syncthreads();  // aBuf reads done; safe to overwrite as hBuf

  // --- residual add, write pre-norm x to hBuf ---
#pragma unroll
  for (int nt = 0; nt < 4; ++nt)
#pragma unroll
    for (int r = 0; r < 8; ++r) {
      int row = r + 8 * hi, col = nt * 16 + lr;
      hBuf[row * HID + col] = acc2[nt][r] + eBuf[row * HID + col];
    }
  __syncthreads();

  // --- layernorm: lane handles (row=lr, cols 32*hi..32*hi+31); pair-combine ---
  {
    float vals[32];
    float s = 0.f, ss = 0.f;
#pragma unroll
    for (int c = 0; c < 32; ++c) {
      float v = hBuf[lr * HID + 32 * hi + c];
      vals[c] = v;
      s += v;
      ss += v * v;
    }
    s += __shfl_xor(s, 16, 32);
    ss += __shfl_xor(ss, 16, 32);
    float mu   = s * (1.f / 64.f);
    float var  = ss * (1.f / 64.f) - mu * mu;
    float rstd = rsqrtf(var + 1e-5f);
#pragma unroll
    for (int c = 0; c < 32; ++c) {
      int col = 32 * hi + c;
      hBuf[lr * HID + col] = (vals[c] - mu) * rstd * gamma[col] + beta[col];
    }
  }
  __syncthreads();

  // --- key projections: h @ Ws + bs, h @ We + be (shared A-fragments) ---
  v8f accs[2], acce[2];
#pragma unroll
  for (int nt = 0; nt < 2; ++nt) {
    float bv1 = bs[nt * 16 + lr], bv2 = be[nt * 16 + lr];
#pragma unroll
    for (int r = 0; r < 8; ++r) {
      accs[nt][r] = bv1;
      acce[nt][r] = bv2;
    }
  }
#pragma unroll
  for (int kc = 0; kc < 16; ++kc) {
    v2f af = *(const v2f*)&hBuf[lr * HID + 4 * kc + 2 * hi];
    const int k2 = 2 * kc + hi;
#pragma unroll
    for (int nt = 0; nt < 2; ++nt) {
      v2f bfs = *(const v2f*)&Wsp[((size_t)k2 * HALFD + nt * 16 + lr) * 2];
      accs[nt] = wmma_f32(af, bfs, accs[nt]);
      v2f bfe = *(const v2f*)&Wep[((size_t)k2 * HALFD + nt * 16 + lr) * 2];
      acce[nt] = wmma_f32(af, bfe, acce[nt]);
    }
  }

  float* ksd = ksb + ((size_t)b * LSEQ + t0) * HALFD;
  float* ked = keb + ((size_t)b * LSEQ + t0) * HALFD;
#pragma unroll
  for (int nt = 0; nt < 2; ++nt)
#pragma unroll
    for (int r = 0; r < 8; ++r) {
      int row = r + 8 * hi, col = nt * 16 + lr;
      ksd[row * HALFD + col] = accs[nt][r];
      ked[row * HALFD + col] = acce[nt][r];
    }
}

// Phase 2: one block per batch; wave0 -> M_s, wave1 -> M_e; lane = matrix row.
__global__ void __launch_bounds__(64)
scan_kernel(const float* __restrict__ ksb, const float* __restrict__ keb,
            const float* __restrict__ Wo, const float* __restrict__ bo,
            float* __restrict__ out) {
  __shared__ float cbuf[64];
  const int b    = blockIdx.x;
  const int tid  = threadIdx.x;
  const int mem  = tid >> 5;  // 0: M_s, 1: M_e
  const int lane = tid & 31;
  const float* kb = (mem == 0 ? ksb : keb) + (size_t)b * LSEQ * HALFD;

  float m[32];
#pragma unroll
  for (int j = 0; j < 32; ++j) m[j] = 0.f;
  const float invL = 1.0f / (float)LSEQ;

  for (int t = 0; t < LSEQ - 1; ++t) {
    const float* kt = kb + t * HALFD;  // wave-uniform address
    float ki = kt[lane];
    float vps = 0.f, d = 1e-6f;
#pragma unroll
    for (int j = 0; j < 32; ++j) {
      float kj = kt[j];
      vps += m[j] * kj;
      d += kj * kj;
    }
    float dv = ki - vps / d;
    if (mem) dv *= (float)(t + 1) * invL;
#pragma unroll
    for (int j = 0; j < 32; ++j) m[j] += dv * kt[j];
  }

  // c = M @ k_{L-1}  (k at t=L-1 equals q@W+b by construction)
  {
    const float* kq = kb + (size_t)(LSEQ - 1) * HALFD;
    float c = 0.f;
#pragma unroll
    for (int j = 0; j < 32; ++j) c += m[j] * kq[j];
    cbuf[mem * 32 + lane] = c;
  }
  __syncthreads();

  // out[b] = concat(cs, ce) @ Wo + bo   (64 threads, one output column each)
  {
    int n = tid;
    float o = bo[n];
#pragma unroll
    for (int i = 0; i < 64; ++i) o += cbuf[i] * Wo[i * 64 + n];
    out[(size_t)b * 64 + n] = o;
  }
}

extern "C" void kernel_launch(void* const* d_in, const int* in_sizes, int n_in,
                              void* d_out, int out_size, void* d_ws,
                              size_t ws_size, hipStream_t stream) {
  const int*   seq   = (const int*)d_in[0];
  const float* embed = (const float*)d_in[1];
  const float* W1    = (const float*)d_in[2];
  const float* b1    = (const float*)d_in[3];
  const float* W2    = (const float*)d_in[4];
  const float* b2    = (const float*)d_in[5];
  const float* gamma = (const float*)d_in[6];
  const float* beta  = (const float*)d_in[7];
  const float* Ws    = (const float*)d_in[8];
  const float* bs    = (const float*)d_in[9];
  const float* We    = (const float*)d_in[10];
  const float* be    = (const float*)d_in[11];
  const float* Wo    = (const float*)d_in[12];
  const float* bo    = (const float*)d_in[13];

  float* ws   = (float*)d_ws;
  const size_t kelems = (size_t)BSZ * LSEQ * HALFD;  // 16.8M floats each
  float* ksb = ws;
  float* keb = ksb + kelems;
  float* W1p = keb + kelems;
  float* W2p = W1p + 64 * 128;
  float* Wsp = W2p + 128 * 64;
  float* Wep = Wsp + 64 * 32;

  // pre-pack weights for single-float2 B-fragment loads
  pack_pairs_kernel<<<(4096 + 255) / 256, 256, 0, stream>>>(W1, W1p, 32, 128);
  pack_pairs_kernel<<<(4096 + 255) / 256, 256, 0, stream>>>(W2, W2p, 64, 64);
  pack_pairs_kernel<<<(1024 + 255) / 256, 256, 0, stream>>>(Ws, Wsp, 32, 32);
  pack_pairs_kernel<<<(1024 + 255) / 256, 256, 0, stream>>>(We, Wep, 32, 32);

  // phase 1: 32768 tiles of 16 tokens, 4 waves (tiles) per block
  ff_tokens_kernel<<<8192, 128, 4 * 3072 * sizeof(float), stream>>>(
      seq, embed, W1p, b1, W2p, b2, gamma, beta, Wsp, bs, Wep, be, ksb, keb);

  // phase 2: sequential scan + fused readout
  scan_kernel<<<BSZ, 64, 0, stream>>>(ksb, keb, Wo, bo, (float*)d_out);
}